// CoLaKG_model_56264071578153
// MI455X (gfx1250) — compile-verified
//
#include <hip/hip_runtime.h>
#include <cmath>

#define NUM_USERS_ 60000
#define NUM_ITEMS_ 30000
#define D_ 64
#define SEM_ 1024
#define NNZ_ 2000000
#define NTOT_ 90000   // NUM_USERS_ + NUM_ITEMS_

typedef float v2f __attribute__((ext_vector_type(2)));
typedef float v8f __attribute__((ext_vector_type(8)));

// ---------------------------------------------------------------------------
// Fused GEMM:  out = (E + elu(A @ W + bias)) * 0.5
// A: M x 1024, W: 1024 x 64, bias: 64, E/out: M x 64
// Block = 256 threads (8 waves). Block tile = 128 rows x 64 cols.
// Each wave computes a 16x64 strip via V_WMMA_F32_16X16X4_F32 (4 C tiles).
//
// W is staged into LDS PRE-SWIZZLED into B-fragment order so each fragment is
// a single lane-contiguous ds_load_b64. The next A chunk is warmed into
// WGP$/L2 with global_prefetch_b8 (__builtin_prefetch) one iteration ahead —
// zero registers, zero counters — so the direct global->LDS staging loads hit
// cache instead of exposing full HBM latency. (A register pipeline was tried
// and spilled to scratch; prefetch achieves the overlap without spills.)
// ---------------------------------------------------------------------------
__global__ __launch_bounds__(256)
void fused_sem_gemm(const float* __restrict__ A,
                    const float* __restrict__ W,
                    const float* __restrict__ bias,
                    const float* __restrict__ E,
                    float* __restrict__ out,
                    int M)
{
    __shared__ __align__(16) float sA [128 * 64];        // 32 KB, natural layout
    __shared__ __align__(16) float sWf[16 * 4 * 32 * 2]; // 16 KB, fragment order

    const int tid  = threadIdx.x;
    const int wave = tid >> 5;
    const int lane = tid & 31;
    const int m    = lane & 15;   // M index of A frag / N index of B frag
    const int hi   = lane >> 4;   // selects K pair {0,1} vs {2,3}
    const int rowBase = blockIdx.x * 128;

    v8f acc[4];
#pragma unroll
    for (int nb = 0; nb < 4; ++nb)
        acc[nb] = (v8f){0.f, 0.f, 0.f, 0.f, 0.f, 0.f, 0.f, 0.f};

    // warm chunk kb = 0 (one lane per 128B cacheline)
    if ((tid & 7) == 0) {
#pragma unroll
        for (int i = 0; i < 8; ++i) {
            int idx = tid + i * 256;
            int gr  = rowBase + (idx >> 4); if (gr >= M) gr = M - 1;
            __builtin_prefetch(A + (size_t)gr * SEM_ + (idx & 15) * 4, 0, 3);
        }
    }

    for (int kb = 0; kb < 16; ++kb) {
        const int k0 = kb * 64;

        // prefetch chunk kb+1 (completes during this iteration's WMMAs)
        if (kb < 15 && (tid & 7) == 0) {
            const int k1 = k0 + 64;
#pragma unroll
            for (int i = 0; i < 8; ++i) {
                int idx = tid + i * 256;
                int gr  = rowBase + (idx >> 4); if (gr >= M) gr = M - 1;
                __builtin_prefetch(A + (size_t)gr * SEM_ + k1 + (idx & 15) * 4, 0, 3);
            }
        }

        // stage A chunk: 128 rows x 64 k-cols (float4-coalesced, cache-hot)
#pragma unroll
        for (int i = 0; i < 8; ++i) {
            int idx = tid + i * 256;          // 0..2047
            int gr  = rowBase + (idx >> 4); if (gr >= M) gr = M - 1;
            ((float4*)sA)[idx] = *(const float4*)(A + (size_t)gr * SEM_ + k0 + (idx & 15) * 4);
        }

        // stage W chunk pre-swizzled into fragment order:
        //   pair p = (ks*4 + nb)*32 + l  ->  { W[k0+4ks+2*(l>>4)  ][16nb + (l&15)],
        //                                     W[k0+4ks+2*(l>>4)+1][16nb + (l&15)] }
#pragma unroll
        for (int i = 0; i < 8; ++i) {
            int p   = tid + i * 256;          // 0..2047
            int ks  = p >> 7;
            int rem = p & 127;
            int nb  = rem >> 5;
            int l   = rem & 31;
            int row = k0 + ks * 4 + (l >> 4) * 2;
            int col = nb * 16 + (l & 15);
            v2f w;
            w.x = W[(size_t)row * D_ + col];
            w.y = W[(size_t)(row + 1) * D_ + col];
            ((v2f*)sWf)[p] = w;
        }
        __syncthreads();

        // 16 k-steps x 4 n-blocks of fp32 WMMA
#pragma unroll
        for (int ks = 0; ks < 16; ++ks) {
            const int kk = ks * 4 + hi * 2;
            v2f a = *(const v2f*)(&sA[(wave * 16 + m) * 64 + kk]);
#pragma unroll
            for (int nb = 0; nb < 4; ++nb) {
                v2f b = ((const v2f*)sWf)[(ks * 4 + nb) * 32 + lane];
                acc[nb] = __builtin_amdgcn_wmma_f32_16x16x4_f32(
                    false, a, false, b, (short)0, acc[nb], false, false);
            }
        }
        __syncthreads();
    }

    // Epilogue: bias + ELU + merge. C layout: VGPR v -> M = v + hi*8, N = lane&15.
#pragma unroll
    for (int nb = 0; nb < 4; ++nb) {
        const int col = nb * 16 + m;
        const float bv = bias[col];
#pragma unroll
        for (int v = 0; v < 8; ++v) {
            const int row = rowBase + wave * 16 + v + hi * 8;
            if (row < M) {
                float x = acc[nb][v] + bv;
                x = (x > 0.f) ? x : expm1f(x);
                out[(size_t)row * D_ + col] = (E[(size_t)row * D_ + col] + x) * 0.5f;
            }
        }
    }
}

// ---------------------------------------------------------------------------
// w12[0:1024]  = W_att @ a_att[:32]   (per-feature fold of the GAT projection)
// w12[1024:..] = W_att @ a_att[32:64]
// ---------------------------------------------------------------------------
__global__ void att_proj(const float* __restrict__ W_att,
                         const float* __restrict__ a_att,
                         float* __restrict__ w12)
{
    int j = blockIdx.x * 256 + threadIdx.x;
    if (j < SEM_) {
        float s1 = 0.f, s2 = 0.f;
#pragma unroll
        for (int h = 0; h < 32; ++h) {
            float w = W_att[j * 32 + h];
            s1 += w * a_att[h];
            s2 += w * a_att[32 + h];
        }
        w12[j]        = s1;
        w12[SEM_ + j] = s2;
    }
}

// ---------------------------------------------------------------------------
// p1[i] = semantic_emb[i,:] . w12[0:1024]; p2[i] = . w12[1024:2048]
// One wave per row; shuffle reduction. Second pass over semantic_emb is
// largely L2-resident (123 MB < 192 MB L2).
// ---------------------------------------------------------------------------
__global__ __launch_bounds__(256)
void sem_matvec(const float* __restrict__ S, const float* __restrict__ w12,
                float* __restrict__ p1, float* __restrict__ p2)
{
    const int wave = threadIdx.x >> 5;
    const int lane = threadIdx.x & 31;
    const int row  = blockIdx.x * 8 + wave;
    if (row >= NUM_ITEMS_) return;
    const float* Sr = S + (size_t)row * SEM_;
    float s1 = 0.f, s2 = 0.f;
#pragma unroll 8
    for (int t = 0; t < 32; ++t) {
        int j = t * 32 + lane;
        float x = Sr[j];
        s1 += x * w12[j];
        s2 += x * w12[SEM_ + j];
    }
#pragma unroll
    for (int off = 16; off > 0; off >>= 1) {
        s1 += __shfl_xor(s1, off);
        s2 += __shfl_xor(s2, off);
    }
    if (lane == 0) { p1[row] = s1; p2[row] = s2; }
}

// ---------------------------------------------------------------------------
// GAT aggregation: att = softmax(leaky_relu(p1[adj] + p2)), then
// items_final = (Mi + elu(sum_k att_k * Mi[adj_k])) * 0.5.
// One wave per item; grid*8 == NUM_ITEMS_ exactly.
// ---------------------------------------------------------------------------
__global__ __launch_bounds__(256)
void gat_aggregate(const int* __restrict__ adj,
                   const float* __restrict__ p1,
                   const float* __restrict__ p2,
                   const float* __restrict__ Mi,
                   float* __restrict__ outItems)
{
    __shared__ int   sIdx[8][32];
    __shared__ float sAtt[8][32];
    const int wave = threadIdx.x >> 5;
    const int lane = threadIdx.x & 31;
    const int item = blockIdx.x * 8 + wave;

    const int nb = adj[(size_t)item * 32 + lane];
    float e = p1[nb] + p2[item];
    e = (e > 0.f) ? e : 0.2f * e;           // leaky_relu(alpha=0.2)
    float mx = e;
#pragma unroll
    for (int off = 16; off > 0; off >>= 1) mx = fmaxf(mx, __shfl_xor(mx, off));
    float ex = expf(e - mx);
    float sum = ex;
#pragma unroll
    for (int off = 16; off > 0; off >>= 1) sum += __shfl_xor(sum, off);
    sIdx[wave][lane] = nb;
    sAtt[wave][lane] = ex / sum;
    __syncthreads();

    float h0 = 0.f, h1 = 0.f;
#pragma unroll 4
    for (int k = 0; k < 32; ++k) {
        const int   r = sIdx[wave][k];
        const float a = sAtt[wave][k];
        h0 += a * Mi[(size_t)r * D_ + lane];
        h1 += a * Mi[(size_t)r * D_ + 32 + lane];
    }
    h0 = (h0 > 0.f) ? h0 : expm1f(h0);
    h1 = (h1 > 0.f) ? h1 : expm1f(h1);
    outItems[(size_t)item * D_ + lane]      = (Mi[(size_t)item * D_ + lane]      + h0) * 0.5f;
    outItems[(size_t)item * D_ + 32 + lane] = (Mi[(size_t)item * D_ + 32 + lane] + h1) * 0.5f;
}

// ---------------------------------------------------------------------------
// Elementwise helpers
// ---------------------------------------------------------------------------
__global__ void fill_zero(float* __restrict__ p, size_t n)
{
    size_t i = (size_t)blockIdx.x * 256 + threadIdx.x;
    if (i < n) p[i] = 0.f;
}
__global__ void copy_buf(float* __restrict__ dst, const float* __restrict__ src, size_t n)
{
    size_t i = (size_t)blockIdx.x * 256 + threadIdx.x;
    if (i < n) dst[i] = src[i];
}
__global__ void add_buf(float* __restrict__ acc, const float* __restrict__ x, size_t n)
{
    size_t i = (size_t)blockIdx.x * 256 + threadIdx.x;
    if (i < n) acc[i] += x[i];
}

// ---------------------------------------------------------------------------
// SpMM layer: next[r,:] += v * prev[c,:] for each edge. One wave per edge,
// 2 dims per lane; f32 atomics land in L2 (23 MB table is L2-resident).
// ---------------------------------------------------------------------------
__global__ __launch_bounds__(256)
void spmm_edges(const int* __restrict__ rows, const int* __restrict__ cols,
                const float* __restrict__ vals,
                const float* __restrict__ prev, float* __restrict__ next)
{
    const int wave = threadIdx.x >> 5;
    const int lane = threadIdx.x & 31;
    const long long e = (long long)blockIdx.x * 8 + wave;
    if (e >= NNZ_) return;
    const int   r = rows[e];
    const int   c = cols[e];
    const float v = vals[e];
    const float* pc = prev + (size_t)c * D_;
    float*       pr = next + (size_t)r * D_;
    atomicAdd(&pr[lane],      v * pc[lane]);
    atomicAdd(&pr[32 + lane], v * pc[32 + lane]);
}

// ---------------------------------------------------------------------------
// gamma[b] = dot(mean_u, mean_i) = dot(acc_u, acc_i) / 16   (mean over 4 embs)
// One wave per batch element; grid*8 == 4096 exactly.
// ---------------------------------------------------------------------------
__global__ __launch_bounds__(256)
void final_dot(const int* __restrict__ users, const int* __restrict__ items,
               const float* __restrict__ acc, float* __restrict__ out)
{
    const int wave = threadIdx.x >> 5;
    const int lane = threadIdx.x & 31;
    const int b = blockIdx.x * 8 + wave;
    const int u  = users[b];
    const int it = items[b];
    const float* pu = acc + (size_t)u * D_;
    const float* pi = acc + (size_t)(NUM_USERS_ + it) * D_;
    float s = pu[lane] * pi[lane] + pu[32 + lane] * pi[32 + lane];
#pragma unroll
    for (int off = 16; off > 0; off >>= 1) s += __shfl_xor(s, off);
    if (lane == 0) out[b] = s * (1.0f / 16.0f);
}

// ---------------------------------------------------------------------------
extern "C" void kernel_launch(void* const* d_in, const int* in_sizes, int n_in,
                              void* d_out, int out_size, void* d_ws, size_t ws_size,
                              hipStream_t stream)
{
    const int*   users  = (const int*)  d_in[0];
    const int*   items  = (const int*)  d_in[1];
    const int*   adj    = (const int*)  d_in[2];
    const int*   grows  = (const int*)  d_in[3];
    const int*   gcols  = (const int*)  d_in[4];
    const float* gvals  = (const float*)d_in[5];
    const float* semE   = (const float*)d_in[6];
    const float* usemE  = (const float*)d_in[7];
    const float* embU   = (const float*)d_in[8];
    const float* embI   = (const float*)d_in[9];
    const float* W_sem  = (const float*)d_in[10];
    const float* b_sem  = (const float*)d_in[11];
    const float* W_usem = (const float*)d_in[12];
    const float* b_usem = (const float*)d_in[13];
    const float* W_att  = (const float*)d_in[14];
    const float* a_att  = (const float*)d_in[15];
    float* out = (float*)d_out;

    // workspace layout (floats)
    float* ws = (float*)d_ws;
    size_t off = 0;
    float* Mi   = ws + off; off += (size_t)NUM_ITEMS_ * D_;   // items_emb_merged (pre-GAT)
    float* p1   = ws + off; off += NUM_ITEMS_;
    float* p2   = ws + off; off += NUM_ITEMS_;
    float* w12  = ws + off; off += 2 * SEM_;
    float* embA = ws + off; off += (size_t)NTOT_ * D_;        // layer ping
    float* embB = ws + off; off += (size_t)NTOT_ * D_;        // layer pong
    float* acc  = ws + off; off += (size_t)NTOT_ * D_;        // sum of embs

    // 1) items merged embedding (WMMA GEMM, fused elu+merge) -> Mi
    fused_sem_gemm<<<235, 256, 0, stream>>>(semE, W_sem, b_sem, embI, Mi, NUM_ITEMS_);
    // 2) users merged embedding -> rows [0, NUM_USERS) of embA
    fused_sem_gemm<<<469, 256, 0, stream>>>(usemE, W_usem, b_usem, embU, embA, NUM_USERS_);
    // 3) fold attention vectors through W_att
    att_proj<<<4, 256, 0, stream>>>(W_att, a_att, w12);
    // 4) per-item attention logit components
    sem_matvec<<<3750, 256, 0, stream>>>(semE, w12, p1, p2);
    // 5) GAT update -> rows [NUM_USERS, NTOT) of embA
    gat_aggregate<<<3750, 256, 0, stream>>>(adj, p1, p2, Mi,
                                            embA + (size_t)NUM_USERS_ * D_);

    const size_t nAll = (size_t)NTOT_ * D_;          // 5,760,000
    const int    gAll = 22500;                       // * 256 == nAll
    copy_buf<<<gAll, 256, 0, stream>>>(acc, embA, nAll);

    float* prev = embA;
    float* next = embB;
    for (int l = 0; l < 3; ++l) {
        fill_zero<<<gAll, 256, 0, stream>>>(next, nAll);
        spmm_edges<<<NNZ_ / 8, 256, 0, stream>>>(grows, gcols, gvals, prev, next);
        add_buf<<<gAll, 256, 0, stream>>>(acc, next, nAll);
        float* t = prev; prev = next; next = t;
    }

    final_dot<<<512, 256, 0, stream>>>(users, items, acc, out);
}